// LinearAttention_18794776887707
// MI455X (gfx1250) — compile-verified
//
#include <hip/hip_runtime.h>
#include <hip/hip_bf16.h>
#include <math.h>

typedef float v8f __attribute__((ext_vector_type(8)));
typedef float v4f __attribute__((ext_vector_type(4)));
typedef float v2f __attribute__((ext_vector_type(2)));
typedef int   v4i __attribute__((ext_vector_type(4)));

#define AS1 __attribute__((address_space(1)))
#define AS3 __attribute__((address_space(3)))

// Problem constants (fixed by setup_inputs)
#define Bsz   8
#define Nsz   4096
#define Csz   512
#define C2    1024
#define HEADS 16
#define DIM   32
#define IMGH  64
#define IMGW  64
#define Mtot  (Bsz * Nsz)          // 32768 rows
#define NCHUNK 8                   // n-split for kv partials

// Workspace layout (float offsets)
#define Q_OFF   0ul
#define K_OFF   16777216ul         // + B*N*C
#define KV_OFF  33554432ul         // + B*N*C          (B*H*32*32  = 131072)
#define KM_OFF  33685504ul         // + B*H*32*32      (B*H*32     = 4096)
#define WT_OFF  33689600ul         // + B*H*32         (C*2C       = 524288)
#define KVP_OFF 34213888ul         // + C*2C           (NCHUNK*B*H*1024 = 1048576)
#define KMP_OFF 35262464ul         // + partial kv     (NCHUNK*B*H*32   = 32768)

static __device__ __forceinline__ v8f wmma_f32(v2f a, v2f b, v8f c) {
    return __builtin_amdgcn_wmma_f32_16x16x4_f32(false, a, false, b, (short)0, c,
                                                 false, false);
}

// ---- CDNA5 async global->LDS DMA (ASYNCcnt) with fallback ------------------
#if defined(__gfx1250__) && \
    __has_builtin(__builtin_amdgcn_global_load_async_to_lds_b128) && \
    __has_builtin(__builtin_amdgcn_global_load_async_to_lds_b32)
#define HAVE_ASYNC 1
#else
#define HAVE_ASYNC 0
#endif

#if HAVE_ASYNC
static __device__ __forceinline__ void async_ld_b128(const float* g, float* l) {
    __builtin_amdgcn_global_load_async_to_lds_b128(
        (AS1 v4i*)(v4i*)(g), (AS3 v4i*)(v4i*)(l), 0, 0);
}
static __device__ __forceinline__ void async_ld_b32(const float* g, float* l) {
    __builtin_amdgcn_global_load_async_to_lds_b32(
        (AS1 int*)(int*)(g), (AS3 int*)(int*)(l), 0, 0);
}
static __device__ __forceinline__ void wait_async0() {
#if __has_builtin(__builtin_amdgcn_s_wait_asynccnt)
    __builtin_amdgcn_s_wait_asynccnt(0);
#else
    asm volatile("s_wait_asynccnt 0x0" ::: "memory");
#endif
}
#else
static __device__ __forceinline__ void wait_async0() {}
#endif

// ---------------------------------------------------------------------------
// Kernel 0: transpose w_qk [1024,512] -> wT [512,1024] for coalesced B loads
// ---------------------------------------------------------------------------
__global__ void k_transpose_w(const float* __restrict__ w, float* __restrict__ wT) {
    int idx = blockIdx.x * 256 + threadIdx.x;      // idx = c*1024 + j
    if (idx < Csz * C2) {
        int c = idx >> 10;
        int j = idx & (C2 - 1);
        wT[idx] = w[j * Csz + c];
    }
}

// ---------------------------------------------------------------------------
// Kernel 1: QK = x @ w^T + b, elu+1, split-store Q and K
// block: 256 thr (8 waves); tile 128 rows x 128 cols; wave 32x64 (8 accums)
// Double-buffered LDS; next tile DMA'd (async-to-LDS) during WMMA compute.
// ---------------------------------------------------------------------------
#define XS_STRIDE 36               // 144B rows: 16B aligned, conflict-free
#define XS_ELEMS  (128 * XS_STRIDE)
#define WP_ELEMS  (16 * 256)       // [kpair][j*2 + (krow&1)]

__global__ void __launch_bounds__(256)
k_qk_gemm(const float* __restrict__ x, const float* __restrict__ wT,
          const float* __restrict__ bqk, float* __restrict__ ws) {
    __shared__ float xs[2][XS_ELEMS];
    __shared__ float wp[2][WP_ELEMS];

    float* Q = ws + Q_OFF;
    float* K = ws + K_OFF;

    const int tid  = threadIdx.x;
    const int lane = tid & 31;
    const int lo   = lane & 15;
    const int half = lane >> 4;
    const int wv   = tid >> 5;           // 0..7
    const int wm   = wv & 3;             // m sub-tile (x32 rows)
    const int wj   = wv >> 2;            // j sub-tile (x64 cols)

    const int m0 = blockIdx.x * 128;
    const int j0 = blockIdx.y * 128;

    v8f acc[2][4];
    for (int s = 0; s < 2; ++s)
        for (int t = 0; t < 4; ++t) acc[s][t] = (v8f)0.0f;

    // -- tile staging: x[128x32] (b128 chunks), w[32x128] K-pair interleaved --
    auto stage = [&](int c0, int buf) {
#if HAVE_ASYNC
        #pragma unroll
        for (int it = 0; it < 4; ++it) {
            int idx = it * 256 + tid;                 // 1024 float4
            int r = idx >> 3, q = idx & 7;
            async_ld_b128(x + (size_t)(m0 + r) * Csz + c0 + q * 4,
                          &xs[buf][r * XS_STRIDE + q * 4]);
        }
        #pragma unroll
        for (int it = 0; it < 16; ++it) {
            int e = it * 256 + tid;                   // 4096 floats
            int r = e >> 7, j = e & 127;
            async_ld_b32(wT + (size_t)(c0 + r) * C2 + j0 + j,
                         &wp[buf][(r >> 1) * 256 + j * 2 + (r & 1)]);
        }
#else
        v4f tx[4], tw[4];
        #pragma unroll
        for (int it = 0; it < 4; ++it) {
            int idx = it * 256 + tid;
            int r = idx >> 3, q = idx & 7;
            tx[it] = *(const v4f*)(x + (size_t)(m0 + r) * Csz + c0 + q * 4);
        }
        #pragma unroll
        for (int it = 0; it < 4; ++it) {
            int idx = it * 256 + tid;
            int r = idx >> 5, q = idx & 31;
            tw[it] = *(const v4f*)(wT + (size_t)(c0 + r) * C2 + j0 + q * 4);
        }
        #pragma unroll
        for (int it = 0; it < 4; ++it) {
            int idx = it * 256 + tid;
            int r = idx >> 3, q = idx & 7;
            *(v4f*)(&xs[buf][r * XS_STRIDE + q * 4]) = tx[it];
        }
        #pragma unroll
        for (int it = 0; it < 4; ++it) {
            int idx = it * 256 + tid;
            int r = idx >> 5, q = idx & 31;
            float* dst = &wp[buf][(r >> 1) * 256 + (r & 1)];
            #pragma unroll
            for (int i = 0; i < 4; ++i) dst[(q * 4 + i) * 2] = tw[it][i];
        }
#endif
    };

    auto compute = [&](int buf) {
        const float* xsb = xs[buf];
        const float* wpb = wp[buf];
        #pragma unroll
        for (int kk = 0; kk < 32; kk += 4) {
            v2f a0 = *(const v2f*)(xsb + (wm * 32 + lo) * XS_STRIDE + kk + half * 2);
            v2f a1 = *(const v2f*)(xsb + (wm * 32 + 16 + lo) * XS_STRIDE + kk + half * 2);
            const float* wrow = wpb + ((kk >> 1) + half) * 256;
            #pragma unroll
            for (int jt = 0; jt < 4; ++jt) {
                v2f bf = *(const v2f*)(wrow + (wj * 64 + jt * 16 + lo) * 2);
                acc[0][jt] = wmma_f32(a0, bf, acc[0][jt]);
                acc[1][jt] = wmma_f32(a1, bf, acc[1][jt]);
            }
        }
    };

    stage(0, 0);
    wait_async0();
    __syncthreads();

    int buf = 0;
    for (int c0 = 0; c0 < Csz; c0 += 32, buf ^= 1) {
        if (c0 + 32 < Csz) stage(c0 + 32, buf ^ 1);   // DMA next tile
        compute(buf);                                  // WMMA on current tile
        wait_async0();                                 // next tile landed
        __syncthreads();
    }

    // epilogue: bias + elu+1, split into Q / K
    for (int jt = 0; jt < 4; ++jt) {
        int j = j0 + wj * 64 + jt * 16 + lo;
        float bias = bqk[j];
        for (int s = 0; s < 2; ++s) {
            for (int r = 0; r < 8; ++r) {
                int mloc = wm * 32 + s * 16 + half * 8 + r;
                size_t m = (size_t)(m0 + mloc);
                float v = acc[s][jt][r] + bias;
                float act = (v > 0.0f) ? (v + 1.0f) : __expf(v);   // elu(v)+1
                if (j < Csz) Q[m * Csz + j]         = act;
                else         K[m * Csz + (j - Csz)] = act;
            }
        }
    }
}

// ---------------------------------------------------------------------------
// Kernel 2: per (b,head,chunk): partial kv = k^T x (32x32), partial k_sum
// grid: (B*H, NCHUNK); block 128 thr (4 waves); wave = one 16x16 quadrant
// ---------------------------------------------------------------------------
__global__ void __launch_bounds__(128)
k_kv_partial(const float* __restrict__ x, float* __restrict__ ws) {
    const float* K = ws + K_OFF;
    float* KVP = ws + KVP_OFF;
    float* KMP = ws + KMP_OFF;

    const int bh = blockIdx.x;           // 0..127
    const int ck = blockIdx.y;           // 0..NCHUNK-1
    const int b  = bh >> 4;
    const int hh = bh & 15;
    const int ch = hh * DIM;

    const int tid  = threadIdx.x;
    const int lane = tid & 31;
    const int lo   = lane & 15;
    const int half = lane >> 4;
    const int wv   = tid >> 5;
    const int dt   = wv & 1;             // d quadrant
    const int et   = wv >> 1;            // e quadrant

    const size_t rowbase = (size_t)b * Nsz;
    const int nlo = ck * (Nsz / NCHUNK);
    const int nhi = nlo + (Nsz / NCHUNK);

    v8f acc = (v8f)0.0f;
    float ksum = 0.0f;

    for (int n0 = nlo; n0 < nhi; n0 += 4) {
        int nA = n0 + half * 2;
        v2f a, bf;
        // A = k^T : A[d][k] ; d = dt*16+lo, k(n) = 2*half + vgpr
        a[0] = K[(rowbase + nA)     * Csz + ch + dt * 16 + lo];
        a[1] = K[(rowbase + nA + 1) * Csz + ch + dt * 16 + lo];
        // B = x   : B[k][e] ; k(n) = 2*half + vgpr, e = et*16+lo
        bf[0] = x[(rowbase + nA)     * Csz + ch + et * 16 + lo];
        bf[1] = x[(rowbase + nA + 1) * Csz + ch + et * 16 + lo];
        acc = wmma_f32(a, bf, acc);
        ksum += a[0] + a[1];             // partial over n for d = dt*16+lo
    }

    // k_sum partial: combine the two n-halves held by lane and lane^16
    float tot = ksum + __shfl_xor(ksum, 16, 32);
    if (et == 0 && lane < 16)
        KMP[((size_t)ck * 128 + bh) * DIM + dt * 16 + lane] = tot;

    // store kv partial quadrant
    for (int r = 0; r < 8; ++r) {
        int drow = dt * 16 + half * 8 + r;
        int e    = et * 16 + lo;
        KVP[((size_t)ck * 128 + bh) * (DIM * DIM) + drow * DIM + e] = acc[r];
    }
}

// ---------------------------------------------------------------------------
// Kernel 2b: deterministic reduction of kv / k_mean partials (no atomics).
// KV is stored K-pair-interleaved: KV[(d>>1)*64 + e*2 + (d&1)] so kernel 3's
// B fragments are single aligned b64 loads.
// ---------------------------------------------------------------------------
__global__ void __launch_bounds__(256)
k_kv_reduce(float* __restrict__ ws) {
    const float* KVP = ws + KVP_OFF;
    const float* KMP = ws + KMP_OFF;
    float* KV = ws + KV_OFF;
    float* KM = ws + KM_OFF;

    const int bh = blockIdx.x;
    const float invN = 1.0f / (float)Nsz;

    for (int idx = threadIdx.x; idx < DIM * DIM; idx += 256) {
        int d = idx >> 5, e = idx & 31;
        float s = 0.0f;
        for (int c = 0; c < NCHUNK; ++c)
            s += KVP[((size_t)c * 128 + bh) * (DIM * DIM) + idx];
        KV[(size_t)bh * (DIM * DIM) + (d >> 1) * 64 + e * 2 + (d & 1)] = s * invN;
    }
    if (threadIdx.x < DIM) {
        float s = 0.0f;
        for (int c = 0; c < NCHUNK; ++c)
            s += KMP[((size_t)c * 128 + bh) * DIM + threadIdx.x];
        KM[(size_t)bh * DIM + threadIdx.x] = s * invN;
    }
}

// ---------------------------------------------------------------------------
// Kernel 3: out = z * (q @ kv) + LePE(x);  z = 1/(q . k_mean + 1e-6)
// grid: (N/64, B*HEADS); block 128 thr (4 waves); wave = 16 rows x 32 cols
// ---------------------------------------------------------------------------
__global__ void __launch_bounds__(128)
k_attn_out(const float* __restrict__ x, const float* __restrict__ wlepe,
           const float* __restrict__ blepe, const float* __restrict__ ws_c,
           float* __restrict__ out) {
    const float* Q  = ws_c + Q_OFF;
    const float* KV = ws_c + KV_OFF;     // K-pair interleaved
    const float* KM = ws_c + KM_OFF;

    const int bh = blockIdx.y;
    const int b  = bh >> 4;
    const int hh = bh & 15;
    const int ch = hh * DIM;

    const int tid  = threadIdx.x;
    const int lane = tid & 31;
    const int lo   = lane & 15;
    const int half = lane >> 4;
    const int wv   = tid >> 5;

    const int n_base = blockIdx.x * 64 + wv * 16;
    const size_t rowbase = (size_t)b * Nsz;
    const size_t kvbase  = (size_t)bh * DIM * DIM;
    const size_t kmbase  = (size_t)bh * DIM;

    v8f acc0 = (v8f)0.0f, acc1 = (v8f)0.0f;
    float zpart = 0.0f;

    #pragma unroll
    for (int dk = 0; dk < DIM; dk += 4) {
        int k0 = dk + half * 2;          // even
        // A = q : row = n_base+lo, k = dk + 2*half + vgpr  (one b64 load)
        v2f a  = *(const v2f*)(Q + (rowbase + n_base + lo) * Csz + ch + k0);
        // B = kv : pair-interleaved -> single b64 per 16-col tile
        v2f b0 = *(const v2f*)(KV + kvbase + (k0 >> 1) * 64 + lo * 2);
        v2f b1 = *(const v2f*)(KV + kvbase + (k0 >> 1) * 64 + (16 + lo) * 2);
        acc0 = wmma_f32(a, b0, acc0);
        acc1 = wmma_f32(a, b1, acc1);
        zpart += a[0] * KM[kmbase + k0] + a[1] * KM[kmbase + k0 + 1];
    }

    // z per row: lane 'lo' (both halves) holds z for row n_base+lo
    float dot = zpart + __shfl_xor(zpart, 16, 32);
    float z   = 1.0f / (dot + 1e-6f);

    for (int r = 0; r < 8; ++r) {
        float zl = __shfl(z, r, 32);
        float zh = __shfl(z, r + 8, 32);
        float zr = half ? zh : zl;
        int n = n_base + half * 8 + r;
        int y  = n >> 6;
        int xp = n & 63;
        float av[2] = {acc0[r], acc1[r]};
        for (int et = 0; et < 2; ++et) {
            int c = ch + et * 16 + lo;
            float lepe = blepe[c];
            #pragma unroll
            for (int dy = 0; dy < 3; ++dy) {
                int yy = y + dy - 1;
                if (yy < 0 || yy >= IMGH) continue;
                #pragma unroll
                for (int dx = 0; dx < 3; ++dx) {
                    int xx = xp + dx - 1;
                    if (xx < 0 || xx >= IMGW) continue;
                    lepe += x[(rowbase + yy * IMGW + xx) * Csz + c] *
                            wlepe[c * 9 + dy * 3 + dx];
                }
            }
            out[(rowbase + n) * Csz + c] = av[et] * zr + lepe;
        }
    }
}

// ---------------------------------------------------------------------------
extern "C" void kernel_launch(void* const* d_in, const int* in_sizes, int n_in,
                              void* d_out, int out_size, void* d_ws, size_t ws_size,
                              hipStream_t stream) {
    (void)in_sizes; (void)n_in; (void)out_size; (void)ws_size;
    const float* x      = (const float*)d_in[0];
    const float* w_qk   = (const float*)d_in[1];
    const float* b_qk   = (const float*)d_in[2];
    const float* w_lepe = (const float*)d_in[3];
    const float* b_lepe = (const float*)d_in[4];
    float* out = (float*)d_out;
    float* ws  = (float*)d_ws;
    float* wT  = ws + WT_OFF;

    k_transpose_w<<<(Csz * C2 + 255) / 256, 256, 0, stream>>>(w_qk, wT);
    k_qk_gemm<<<dim3(Mtot / 128, C2 / 128), 256, 0, stream>>>(x, wT, b_qk, ws);
    k_kv_partial<<<dim3(Bsz * HEADS, NCHUNK), 128, 0, stream>>>(x, ws);
    k_kv_reduce<<<Bsz * HEADS, 256, 0, stream>>>(ws);
    k_attn_out<<<dim3(Nsz / 64, Bsz * HEADS), 128, 0, stream>>>(
        x, w_lepe, b_lepe, ws, out);
}